// RSDPF_35356170781396
// MI455X (gfx1250) — compile-verified
//
#include <hip/hip_runtime.h>
#include <stdint.h>

#define NB 32
#define NT_T 64
#define NM 8
#define NP 4096
#define THREADS 256
#define PPT (NP / THREADS)          // 16 particles per thread
#define LOG2PI 1.8378770664093453f

typedef float v2f __attribute__((ext_vector_type(2)));
typedef float v8f __attribute__((ext_vector_type(8)));

// ---------- stateless counter RNG ----------
__device__ __forceinline__ unsigned pcg(unsigned x) {
    x = x * 747796405u + 2891336453u;
    unsigned w = ((x >> ((x >> 28) + 4u)) ^ x) * 277803737u;
    return (w >> 22) ^ w;
}
__device__ __forceinline__ unsigned mixh(unsigned b, unsigned p, unsigned t, unsigned s) {
    return pcg(p ^ pcg(t ^ pcg(b ^ (s * 0x9E3779B9u))));
}
__device__ __forceinline__ float u01(unsigned h) {
    return ((float)(h >> 8) + 0.5f) * (1.0f / 16777216.0f);   // (0,1)
}

// ---------- wave32 reductions ----------
__device__ __forceinline__ float wredsum(float v) {
#pragma unroll
    for (int d = 16; d >= 1; d >>= 1) v += __shfl_xor(v, d, 32);
    return v;
}
__device__ __forceinline__ float wredmax(float v) {
#pragma unroll
    for (int d = 16; d >= 1; d >>= 1) v = fmaxf(v, __shfl_xor(v, d, 32));
    return v;
}

// =====================================================================
// K1: persistent particle filter. One block per batch element.
// State (s, m, w, cw) lives in LDS for all 63 sequential steps.
// Weight cumsum uses V_WMMA_F32_16X16X4_F32 tile scans.
// =====================================================================
__global__ void __launch_bounds__(THREADS)
pf_filter(const float* __restrict__ o, const float* __restrict__ matP,
          const float* __restrict__ cA, const float* __restrict__ cB,
          const float* __restrict__ cC, const float* __restrict__ cD,
          const float* __restrict__ sigu, const float* __restrict__ sigv,
          float* __restrict__ ws_s, unsigned char* __restrict__ ws_m,
          float* __restrict__ ws_w, unsigned short* __restrict__ ws_idx)
{
    __shared__ float sh_s[NP];            // 16 KB
    __shared__ float sh_w[NP];            // 16 KB
    __shared__ float sh_cw[NP];           // 16 KB (logw temp, then cumsum)
    __shared__ unsigned char sh_m[NP];    //  4 KB
    __shared__ float sh_Pc[NM * NM];      // cumulative transition rows
    __shared__ float sh_co[4 * NM];       // A,B,C,D coefficients
    __shared__ float redA[8], redB[8];
    __shared__ float sh_scan[8];

    const int b    = blockIdx.x;
    const int tid  = threadIdx.x;
    const int wave = tid >> 5;
    const int lane = tid & 31;
    const float su = sigu[0];
    const float sv = sigv[0];
    const float inv_np = 1.0f / (float)NP;

    if (tid < NM) {
        float acc = 0.0f;
        for (int k = 0; k < NM; ++k) { acc += matP[tid * NM + k]; sh_Pc[tid * NM + k] = acc; }
        sh_co[tid] = cA[tid]; sh_co[NM + tid] = cB[tid];
        sh_co[2 * NM + tid] = cC[tid]; sh_co[3 * NM + tid] = cD[tid];
    }

    // ----- t = 0 initialization -----
    {
        size_t cb = ((size_t)b * NT_T + 0) * NP;
#pragma unroll
        for (int k = 0; k < PPT; ++k) {
            int j = tid * PPT + k;
            int m0 = (int)(mixh(b, j, 0, 11) & 7u);
            float s0 = u01(mixh(b, j, 0, 12)) - 0.5f;
            sh_m[j] = (unsigned char)m0; sh_s[j] = s0; sh_w[j] = inv_np;
            ws_m[cb + j] = (unsigned char)m0; ws_s[cb + j] = s0; ws_w[cb + j] = inv_np;
        }
    }
    __syncthreads();

    for (int t = 1; t < NT_T; ++t) {
        const float ot = o[b * NT_T + t];
        const size_t cb = ((size_t)b * NT_T + t) * NP;

        // ----- propagate + log-likelihood -----
        float lmax = -3.4e38f;
#pragma unroll
        for (int k = 0; k < PPT; ++k) {
            int j = tid * PPT + k;
            int mp = sh_m[j];
            float sp = sh_s[j];
            float wp = sh_w[j];
            float u = u01(mixh(b, j, t, 1));
            int mt = NM - 1;
#pragma unroll
            for (int q = 0; q < NM; ++q) { if (u < sh_Pc[mp * NM + q]) { mt = q; break; } }
            float u1 = u01(mixh(b, j, t, 2));
            float u2 = u01(mixh(b, j, t, 3));
            float nrm = sqrtf(-2.0f * __logf(u1)) * __cosf(6.28318530718f * u2);
            float st = sh_co[mt] * sp + sh_co[NM + mt] + nrm * su;
            if (st == 0.0f) st += 1e-4f;
            float mean = sh_co[2 * NM + mt] * sqrtf(fabsf(st)) + sh_co[3 * NM + mt];
            float dd = (ot - mean) / sv;
            float ll = -0.5f * dd * dd - __logf(sv) - 0.5f * LOG2PI;
            float lw = __logf(wp) + ll;
            sh_s[j] = st; sh_m[j] = (unsigned char)mt; sh_cw[j] = lw;
            lmax = fmaxf(lmax, lw);
        }
        // ----- block max (softmax stabilizer) -----
        float M = wredmax(lmax);
        if (lane == 0) redA[wave] = M;
        __syncthreads();
        if (wave == 0) { float x = (lane < 8) ? redA[lane] : -3.4e38f; x = wredmax(x); if (lane == 0) redA[0] = x; }
        __syncthreads();
        M = redA[0];

        // ----- exp, sum, sum-of-squares -----
        float lsum = 0.0f, lsq = 0.0f;
#pragma unroll
        for (int k = 0; k < PPT; ++k) {
            int j = tid * PPT + k;
            float wu = __expf(sh_cw[j] - M);
            sh_w[j] = wu; lsum += wu; lsq += wu * wu;
        }
        float S = wredsum(lsum), Q = wredsum(lsq);
        __syncthreads();
        if (lane == 0) { redA[wave] = S; redB[wave] = Q; }
        __syncthreads();
        if (wave == 0) {
            float xs = (lane < 8) ? redA[lane] : 0.0f;
            float xq = (lane < 8) ? redB[lane] : 0.0f;
            xs = wredsum(xs); xq = wredsum(xq);
            if (lane == 0) { redA[0] = xs; redB[0] = xq; }
        }
        __syncthreads();
        S = redA[0]; Q = redB[0];
        const float invS = 1.0f / S;
        // ESS = S^2/Q (un-normalized);  resample when ESS < N_p
        const bool ess_small = (S * S) < ((float)NP) * Q;

        // ----- normalize; stream raw per-step columns to global (b,t,p) -----
#pragma unroll
        for (int k = 0; k < PPT; ++k) {
            int j = tid * PPT + k;
            float w = sh_w[j] * invS;
            sh_w[j] = w;
            ws_s[cb + j] = sh_s[j];
            ws_m[cb + j] = sh_m[j];
            ws_w[cb + j] = ess_small ? inv_np : w;
        }
        __syncthreads();

        // ----- 4096-wide inclusive cumsum of sh_w -> sh_cw via WMMA tile scan ----
        {
            const int half = lane >> 4, l = lane & 15;
            for (int c = wave * 2; c < wave * 2 + 2; ++c) {   // 2 tiles of 256 per wave
                const int base = c * 256;
                v8f acc = {0.f, 0.f, 0.f, 0.f, 0.f, 0.f, 0.f, 0.f};
#pragma unroll
                for (int q = 0; q < 4; ++q) {                 // K in chunks of 4
                    const int k0 = 4 * q + 2 * half;
                    v2f a;                                    // A = X (16x4 slice), ISA A layout
                    a.x = sh_w[base + l * 16 + k0];
                    a.y = sh_w[base + l * 16 + k0 + 1];
                    v2f bb;                                   // B = upper-tri ones slice
                    bb.x = (l >= k0)     ? 1.0f : 0.0f;
                    bb.y = (l >= k0 + 1) ? 1.0f : 0.0f;
                    acc = __builtin_amdgcn_wmma_f32_16x16x4_f32(
                        false, a, false, bb, (short)0, acc, false, false);
                }
#pragma unroll
                for (int r = 0; r < 8; ++r)                   // D layout: vgpr r -> rows r, r+8
                    sh_cw[base + (r + 8 * half) * 16 + l] = acc[r];
            }
            __syncthreads();
            // cross-row offsets: scan the 256 row totals (one per thread)
            float rt = sh_cw[tid * 16 + 15];
            float v = rt;
#pragma unroll
            for (int d = 1; d < 32; d <<= 1) { float up = __shfl_up(v, d, 32); if (lane >= d) v += up; }
            if (lane == 31) sh_scan[wave] = v;
            __syncthreads();
            if (wave == 0 && lane < 8) {
                float x = sh_scan[lane];
#pragma unroll
                for (int d = 1; d < 8; d <<= 1) { float up = __shfl_up(x, d, 8); if ((lane & 7) >= d) x += up; }
                sh_scan[lane] = x;
            }
            __syncthreads();
            float off = ((wave > 0) ? sh_scan[wave - 1] : 0.0f) + (v - rt);
#pragma unroll
            for (int k = 0; k < 16; ++k) sh_cw[tid * 16 + k] += off;
            __syncthreads();
        }

        // ----- resample (binary search in cumsum), record ancestor indices -----
        int idxs[PPT];
#pragma unroll
        for (int k = 0; k < PPT; ++k) {
            int j = tid * PPT + k;
            int id = j;
            if (ess_small) {
                float u2 = u01(mixh(b, j, t, 4));
                int lo = 0, hi = NP;
                while (lo < hi) { int mid = (lo + hi) >> 1; if (sh_cw[mid] < u2) lo = mid + 1; else hi = mid; }
                id = (lo < NP - 1) ? lo : (NP - 1);
            }
            idxs[k] = id;
            ws_idx[cb + j] = (unsigned short)id;
        }
        // gather current state through registers (no extra LDS buffers needed)
        float gs[PPT]; unsigned char gm[PPT];
#pragma unroll
        for (int k = 0; k < PPT; ++k) { gs[k] = sh_s[idxs[k]]; gm[k] = sh_m[idxs[k]]; }
        __syncthreads();
#pragma unroll
        for (int k = 0; k < PPT; ++k) {
            int j = tid * PPT + k;
            sh_s[j] = gs[k]; sh_m[j] = gm[k];
            if (ess_small) sh_w[j] = inv_np;     // w_re; w is NOT reordered (matches reference)
        }
        __syncthreads();
    }
}

// =====================================================================
// K2: ancestor composition + trajectory reconstruction (backward in t),
// in-place gather of raw columns via LDS staging; also computes s_est.
// =====================================================================
__global__ void __launch_bounds__(THREADS)
pf_recon(const unsigned short* __restrict__ ws_idx,
         unsigned char* __restrict__ ws_m, float* __restrict__ ws_s,
         const float* __restrict__ ws_w, float* __restrict__ s_est)
{
    __shared__ unsigned short comp[NP];   // 8 KB
    __shared__ unsigned short idxc[NP];   // 8 KB
    __shared__ float stg_s[NP];           // 16 KB
    __shared__ unsigned char stg_m[NP];   // 4 KB
    __shared__ float red[8];

    const int b = blockIdx.x, tid = threadIdx.x;
    const int wave = tid >> 5, lane = tid & 31;

#pragma unroll
    for (int k = 0; k < PPT; ++k) comp[tid * PPT + k] = (unsigned short)(tid * PPT + k);
    __syncthreads();

    for (int t = NT_T - 1; t >= 0; --t) {
        const size_t cb = ((size_t)b * NT_T + t) * NP;
        if (t > 0) {
#pragma unroll
            for (int k = 0; k < PPT; ++k) { int j = tid * PPT + k; idxc[j] = ws_idx[cb + j]; }
            __syncthreads();
#pragma unroll
            for (int k = 0; k < PPT; ++k) { int j = tid * PPT + k; comp[j] = idxc[comp[j]]; }
        }
#pragma unroll
        for (int k = 0; k < PPT; ++k) {
            int j = tid * PPT + k;
            stg_s[j] = ws_s[cb + j]; stg_m[j] = ws_m[cb + j];
        }
        __syncthreads();
        float dot = 0.0f;
        float gsv[PPT]; unsigned char gmv[PPT];
#pragma unroll
        for (int k = 0; k < PPT; ++k) {
            int j = tid * PPT + k;
            int cj = comp[j];
            gsv[k] = stg_s[cj]; gmv[k] = stg_m[cj];
            dot += ws_w[cb + j] * gsv[k];
        }
#pragma unroll
        for (int k = 0; k < PPT; ++k) {
            int j = tid * PPT + k;
            ws_s[cb + j] = gsv[k]; ws_m[cb + j] = gmv[k];
        }
        dot = wredsum(dot);
        if (lane == 0) red[wave] = dot;
        __syncthreads();
        if (wave == 0) {
            float x = (lane < 8) ? red[lane] : 0.0f;
            x = wredsum(x);
            if (lane == 0) s_est[b * NT_T + t] = x;
        }
        __syncthreads();   // staging buffers free for next column
    }
}

// =====================================================================
// K3: coalesced (b,t,p) -> (b,p,t) transpose into final output slabs.
// z = 0: m (u8 -> int32), z = 1: s (f32), z = 2: w (f32)
// =====================================================================
__global__ void __launch_bounds__(256)
pf_transpose(const unsigned char* __restrict__ ws_m, const float* __restrict__ ws_s,
             const float* __restrict__ ws_w,
             int* __restrict__ m_hist, float* __restrict__ s_hist, float* __restrict__ w_hist)
{
    __shared__ float tile[64][65];
    const int p0 = blockIdx.x * 64, b = blockIdx.y, z = blockIdx.z;
    const int tx = threadIdx.x & 63, ty = threadIdx.x >> 6;
#pragma unroll
    for (int r = 0; r < 16; ++r) {
        int t = ty * 16 + r;
        size_t src = ((size_t)b * NT_T + t) * NP + p0 + tx;
        float v;
        if (z == 0)      v = (float)ws_m[src];
        else if (z == 1) v = ws_s[src];
        else             v = ws_w[src];
        tile[t][tx] = v;
    }
    __syncthreads();
#pragma unroll
    for (int r = 0; r < 16; ++r) {
        int p = ty * 16 + r;
        size_t dst = ((size_t)b * NP + p0 + p) * NT_T + tx;
        float v = tile[tx][p];
        if (z == 0)      m_hist[dst] = (int)v;
        else if (z == 1) s_hist[dst] = v;
        else             w_hist[dst] = v;
    }
}

extern "C" void kernel_launch(void* const* d_in, const int* in_sizes, int n_in,
                              void* d_out, int out_size, void* d_ws, size_t ws_size,
                              hipStream_t stream)
{
    (void)in_sizes; (void)n_in; (void)out_size; (void)ws_size;
    const float* o    = (const float*)d_in[0];
    const float* matP = (const float*)d_in[1];
    const float* cA   = (const float*)d_in[2];
    const float* cB   = (const float*)d_in[3];
    const float* cC   = (const float*)d_in[4];
    const float* cD   = (const float*)d_in[5];
    const float* sigu = (const float*)d_in[6];
    const float* sigv = (const float*)d_in[7];

    float* out    = (float*)d_out;
    float* s_est  = out;                                        // B*T floats
    int*   m_hist = (int*)out + (size_t)NB * NT_T;              // B*NP*T int32
    float* s_hist = out + (size_t)NB * NT_T + (size_t)NB * NP * NT_T;
    float* w_hist = s_hist + (size_t)NB * NP * NT_T;

    char* ws = (char*)d_ws;
    const size_t n = (size_t)NB * NT_T * NP;                    // 8,388,608
    float*          ws_s   = (float*)ws;                        // 32 MB
    float*          ws_w   = (float*)(ws + n * 4);              // 32 MB
    unsigned short* ws_idx = (unsigned short*)(ws + n * 8);     // 16 MB
    unsigned char*  ws_m   = (unsigned char*)(ws + n * 10);     //  8 MB

    pf_filter<<<NB, THREADS, 0, stream>>>(o, matP, cA, cB, cC, cD, sigu, sigv,
                                          ws_s, ws_m, ws_w, ws_idx);
    pf_recon<<<NB, THREADS, 0, stream>>>(ws_idx, ws_m, ws_s, ws_w, s_est);
    dim3 g3(NP / 64, NB, 3);
    pf_transpose<<<g3, 256, 0, stream>>>(ws_m, ws_s, ws_w, m_hist, s_hist, w_hist);
}